// SimpleMoELayer_85143431675949
// MI455X (gfx1250) — compile-verified
//
#include <hip/hip_runtime.h>
#include <hip/hip_bf16.h>

typedef float v2f __attribute__((ext_vector_type(2)));
typedef float v8f __attribute__((ext_vector_type(8)));

#define TOKENS 16384      // B*S = 8*2048
#define HDIM   512
#define IDIM   2048
#define NEXP   8
#define RBLOCKS 64        // TOKENS / 256
#define TM     32         // tokens per block (2 row-tiles of 16)
#define CHUNK  256
#define A_STRIDE 516      // 512 + 4 pad -> conflict-free (stride%64 == 4)
#define H_STRIDE 260      // 256 + 4 pad

// branchless gelu (tanh approx, matches jax.nn.gelu approximate=True)
__device__ __forceinline__ float gelu_tanh(float v) {
    const float c0 = 0.7978845608028654f;   // sqrt(2/pi)
    float u = c0 * (v + 0.044715f * v * v * v);
    float t = __expf(2.0f * u);             // v_exp_f32, no branches
    float th = 1.0f - 2.0f / (t + 1.0f);    // tanh(u), saturates correctly at +-inf
    return 0.5f * v * (1.0f + th);
}

__global__ void moe_init_kernel(int* __restrict__ cnt) {
    if (threadIdx.x < 2 * NEXP) cnt[threadIdx.x] = 0;
}

__global__ __launch_bounds__(256) void moe_router_kernel(
    const float* __restrict__ x, const float* __restrict__ Wg,
    const float* __restrict__ bg,
    int* __restrict__ cnt, int* __restrict__ bucket,
    float* __restrict__ wt, float* __restrict__ Ppart)
{
    __shared__ float wgl[HDIM * NEXP];   // 16 KB, [k][e]
    __shared__ float red[256];

    const int tid = threadIdx.x;
    for (int i = tid; i < HDIM * NEXP; i += 256) wgl[i] = Wg[i];
    __syncthreads();

    const int t = blockIdx.x * 256 + tid;

    float logit[NEXP];
    #pragma unroll
    for (int e = 0; e < NEXP; ++e) logit[e] = bg[e];

    const float4* xr = (const float4*)(x + (size_t)t * HDIM);
    for (int j = 0; j < HDIM / 4; ++j) {
        float4 v = xr[j];
        int k = j * 4;
        #pragma unroll
        for (int e = 0; e < NEXP; ++e) {
            logit[e] += v.x * wgl[(k + 0) * NEXP + e] + v.y * wgl[(k + 1) * NEXP + e]
                      + v.z * wgl[(k + 2) * NEXP + e] + v.w * wgl[(k + 3) * NEXP + e];
        }
    }

    // softmax over 8
    float m = logit[0];
    #pragma unroll
    for (int e = 1; e < NEXP; ++e) m = fmaxf(m, logit[e]);
    float p[NEXP], s = 0.f;
    #pragma unroll
    for (int e = 0; e < NEXP; ++e) { p[e] = expf(logit[e] - m); s += p[e]; }
    float inv = 1.f / s;
    #pragma unroll
    for (int e = 0; e < NEXP; ++e) p[e] *= inv;

    // top-2 (first index wins ties, matching top_k)
    int i0 = 0; float p0 = p[0];
    #pragma unroll
    for (int e = 1; e < NEXP; ++e) if (p[e] > p0) { p0 = p[e]; i0 = e; }
    int i1 = -1; float p1 = -1.f;
    #pragma unroll
    for (int e = 0; e < NEXP; ++e) if (e != i0 && p[e] > p1) { p1 = p[e]; i1 = e; }
    float rn = 1.f / (p0 + p1);

    int j0 = atomicAdd(&cnt[i0], 1);
    bucket[i0 * TOKENS + j0] = t;
    int j1 = atomicAdd(&cnt[NEXP + i1], 1);
    bucket[(NEXP + i1) * TOKENS + j1] = t;
    wt[t]          = p0 * rn;
    wt[TOKENS + t] = p1 * rn;

    // deterministic per-block sum of probs for balance loss
    for (int e = 0; e < NEXP; ++e) {
        __syncthreads();
        red[tid] = p[e];
        __syncthreads();
        for (int off = 128; off > 0; off >>= 1) {
            if (tid < off) red[tid] += red[tid + off];
            __syncthreads();
        }
        if (tid == 0) Ppart[blockIdx.x * NEXP + e] = red[0];
    }
}

__global__ void moe_loss_kernel(const int* __restrict__ cnt,
                                const float* __restrict__ Ppart,
                                float* __restrict__ out_loss)
{
    if (threadIdx.x != 0 || blockIdx.x != 0) return;
    float acc = 0.f;
    for (int e = 0; e < NEXP; ++e) {
        float Pe = 0.f;
        for (int b = 0; b < RBLOCKS; ++b) Pe += Ppart[b * NEXP + e];
        Pe /= (float)TOKENS;
        float fe = (float)(cnt[e] + cnt[NEXP + e]) / (float)(TOKENS * 2);
        acc += fe * Pe;
    }
    *out_loss = 0.01f * (float)NEXP * acc;
}

__global__ __launch_bounds__(256) void moe_expert_kernel(
    const float* __restrict__ x,
    const float* __restrict__ W1, const float* __restrict__ bias1,
    const float* __restrict__ W2, const float* __restrict__ bias2,
    const int* __restrict__ cnt, const int* __restrict__ bucket,
    const float* __restrict__ wt, float* __restrict__ out, int phase)
{
    __shared__ float Alds[TM * A_STRIDE];   // 64.5 KB : x tile 32 x 512
    __shared__ float Hlds[TM * H_STRIDE];   // 32.5 KB : gelu(h) chunk 32 x 256
    __shared__ int   tokLds[TM];
    __shared__ float wtLds[TM];

    const int e = blockIdx.y;
    const int n = cnt[phase * NEXP + e];
    const int row0 = blockIdx.x * TM;
    if (row0 >= n) return;
    const int rows = min(TM, n - row0);

    const int tid = threadIdx.x;
    if (tid < TM) {
        int idx = row0 + ((tid < rows) ? tid : 0);   // pad by replicating row 0
        int tk = bucket[(phase * NEXP + e) * TOKENS + idx];
        tokLds[tid] = tk;
        wtLds[tid] = wt[phase * TOKENS + tk];
    }
    __syncthreads();

    // cooperative load of A tile (32 tokens x 512) into LDS; 8 threads/row
    {
        int mrow = tid >> 3;          // 0..31
        int l = tid & 7;
        const float4* src = (const float4*)(x + (size_t)tokLds[mrow] * HDIM);
        #pragma unroll
        for (int j = 0; j < 16; ++j) {
            int c4 = l + j * 8;       // float4 index 0..127
            float4 v = src[c4];
            float* dst = &Alds[mrow * A_STRIDE + c4 * 4];
            dst[0] = v.x; dst[1] = v.y; dst[2] = v.z; dst[3] = v.w;
        }
    }
    __syncthreads();

    const int wave = tid >> 5;
    const int lane = tid & 31;
    const int l16  = lane & 15;
    const int koff = (lane < 16) ? 0 : 2;   // fp32 WMMA A/B K-pair per lane-half

    const v8f vzero = {};
    v8f acc2[4][2] = {{vzero, vzero}, {vzero, vzero}, {vzero, vzero}, {vzero, vzero}};

    const size_t w1base = (size_t)e * HDIM * IDIM;
    const size_t w2base = (size_t)e * IDIM * HDIM;

    for (int c = 0; c < IDIM / CHUNK; ++c) {
        // ---- GEMM 1 (K-outer): h = gelu(x @ W1 + b1) for this chunk of I cols ----
        {
            const int ncol0 = c * CHUNK + wave * 32 + l16;   // sub-0 global I column
            v8f h00 = {}, h10 = {}, h01 = {}, h11 = {};      // [rowtile][sub]
            const float* bp = W1 + w1base + (size_t)koff * IDIM + ncol0;
            const float* ap = &Alds[l16 * A_STRIDE + koff];
            if (c + 1 < IDIM / CHUNK)
                __builtin_prefetch(bp + CHUNK, 0, 1);        // warm next chunk (L2)
            for (int k = 0; k < HDIM; k += 4) {
                v2f a0, a1, bf0, bf1;
                a0.x = ap[k];
                a0.y = ap[k + 1];
                a1.x = ap[16 * A_STRIDE + k];
                a1.y = ap[16 * A_STRIDE + k + 1];
                bf0.x = bp[0];
                bf0.y = bp[IDIM];
                bf1.x = bp[16];
                bf1.y = bp[IDIM + 16];
                bp += 4 * IDIM;
                // 2 A-pairs + 2 B-pairs feed 4 independent WMMAs
                h00 = __builtin_amdgcn_wmma_f32_16x16x4_f32(false, a0, false, bf0, (short)0, h00, false, false);
                h10 = __builtin_amdgcn_wmma_f32_16x16x4_f32(false, a1, false, bf0, (short)0, h10, false, false);
                h01 = __builtin_amdgcn_wmma_f32_16x16x4_f32(false, a0, false, bf1, (short)0, h01, false, false);
                h11 = __builtin_amdgcn_wmma_f32_16x16x4_f32(false, a1, false, bf1, (short)0, h11, false, false);
            }
            float bv0 = bias1[(size_t)e * IDIM + ncol0];
            float bv1 = bias1[(size_t)e * IDIM + ncol0 + 16];
            int cb = wave * 32 + l16;                        // column within chunk
            #pragma unroll
            for (int r = 0; r < 8; ++r) {
                int row = r + ((lane < 16) ? 0 : 8);
                Hlds[row * H_STRIDE + cb]             = gelu_tanh(h00[r] + bv0);
                Hlds[(16 + row) * H_STRIDE + cb]      = gelu_tanh(h10[r] + bv0);
                Hlds[row * H_STRIDE + cb + 16]        = gelu_tanh(h01[r] + bv1);
                Hlds[(16 + row) * H_STRIDE + cb + 16] = gelu_tanh(h11[r] + bv1);
            }
        }
        __syncthreads();

        // ---- GEMM 2 (K-outer): y += h_chunk @ W2_chunk ----
        {
            const int ncol0 = wave * 64 + l16;               // sub-0 output H column
            const float* bp = W2 + w2base + (size_t)(c * CHUNK + koff) * HDIM + ncol0;
            const float* hp = &Hlds[l16 * H_STRIDE + koff];
            for (int k = 0; k < CHUNK; k += 4) {
                v2f a0, a1;
                a0.x = hp[k];
                a0.y = hp[k + 1];
                a1.x = hp[16 * H_STRIDE + k];
                a1.y = hp[16 * H_STRIDE + k + 1];
                #pragma unroll
                for (int s2 = 0; s2 < 4; ++s2) {
                    v2f b;
                    b.x = bp[16 * s2];
                    b.y = bp[HDIM + 16 * s2];
                    // one A-pair pair feeds all 4 column tiles (8 WMMAs / K-step)
                    acc2[s2][0] = __builtin_amdgcn_wmma_f32_16x16x4_f32(false, a0, false, b, (short)0, acc2[s2][0], false, false);
                    acc2[s2][1] = __builtin_amdgcn_wmma_f32_16x16x4_f32(false, a1, false, b, (short)0, acc2[s2][1], false, false);
                }
                bp += 4 * HDIM;
            }
        }
        __syncthreads();
    }

    // ---- finalize: out = x + w0*y (phase 0)  /  out += w1*y (phase 1) ----
    #pragma unroll
    for (int s2 = 0; s2 < 4; ++s2) {
        int ncol = wave * 64 + s2 * 16 + l16;
        float bias = bias2[(size_t)e * HDIM + ncol];
        #pragma unroll
        for (int rt = 0; rt < 2; ++rt) {
            #pragma unroll
            for (int r = 0; r < 8; ++r) {
                int row = rt * 16 + r + ((lane < 16) ? 0 : 8);
                if (row < rows) {
                    int t = tokLds[row];
                    float y = (acc2[s2][rt][r] + bias) * wtLds[row];
                    float* o = out + (size_t)t * HDIM + ncol;
                    if (phase == 0) {
                        *o = Alds[row * A_STRIDE + ncol] + y;   // x tile resident in LDS
                    } else {
                        *o += y;
                    }
                }
            }
        }
    }
}

extern "C" void kernel_launch(void* const* d_in, const int* in_sizes, int n_in,
                              void* d_out, int out_size, void* d_ws, size_t ws_size,
                              hipStream_t stream) {
    const float* x  = (const float*)d_in[0];
    const float* Wg = (const float*)d_in[1];
    const float* bg = (const float*)d_in[2];
    const float* W1 = (const float*)d_in[3];
    const float* b1 = (const float*)d_in[4];
    const float* W2 = (const float*)d_in[5];
    const float* b2 = (const float*)d_in[6];
    float* out = (float*)d_out;

    // workspace layout (~1.2 MB)
    int*   cnt    = (int*)d_ws;                        // 2*8 ints
    int*   bucket = cnt + 2 * NEXP;                    // 2*8*TOKENS ints
    float* wt     = (float*)(bucket + 2 * NEXP * TOKENS);  // 2*TOKENS floats
    float* Ppart  = wt + 2 * TOKENS;                   // RBLOCKS*8 floats

    hipLaunchKernelGGL(moe_init_kernel, dim3(1), dim3(32), 0, stream, cnt);
    hipLaunchKernelGGL(moe_router_kernel, dim3(RBLOCKS), dim3(256), 0, stream,
                       x, Wg, bg, cnt, bucket, wt, Ppart);
    hipLaunchKernelGGL(moe_loss_kernel, dim3(1), dim3(1), 0, stream,
                       cnt, Ppart, out + (size_t)TOKENS * HDIM);
    hipLaunchKernelGGL(moe_expert_kernel, dim3(TOKENS / TM, NEXP), dim3(256), 0, stream,
                       x, W1, b1, W2, b2, cnt, bucket, wt, out, 0);
    hipLaunchKernelGGL(moe_expert_kernel, dim3(TOKENS / TM, NEXP), dim3(256), 0, stream,
                       x, W1, b1, W2, b2, cnt, bucket, wt, out, 1);
}